// TorchCombineModule_27779848470601
// MI455X (gfx1250) — compile-verified
//
#include <hip/hip_runtime.h>
#include <hip/hip_bf16.h>
#include <cstdint>

// Problem constants (match reference).
#define NUM_CHIPS            8
#define EXPERTS_PER_CHIP     8
#define NUM_EXPERTS_PER_TOK  8
#define SEQ_LEN_PER_CHIP     2048
#define MAX_TOK              2048
#define HIDDEN               2048

typedef uint32_t v4u __attribute__((ext_vector_type(4)));

// ---------------------------------------------------------------------------
// Scatter pass: one 256-thread block per source row (c,e,t).
//   - counter/metadata addresses are uniform per block -> scalar loads +
//     scalar early-exit branch for invalid rows (~50% of blocks do nothing).
//   - each thread: one non-temporal 128-bit load (8 bf16) of the source row,
//     then 4x global_atomic_pk_add_bf16 (CDNA5 packed-bf16 L2 atomic,
//     fire-and-forget / STOREcnt-tracked) sharing one 64-bit address with
//     immediate offsets 0/4/8/12.
// ---------------------------------------------------------------------------
__global__ void __launch_bounds__(256) combine_scatter(
    const __hip_bfloat16* __restrict__ disp,   // (C,E,T,H) bf16
    const int*            __restrict__ meta,   // (C,E,T,3) int32: chip,tok,kk
    const int*            __restrict__ counter,// (C,E) int32
    __hip_bfloat16*       __restrict__ out)    // (C,SEQ,K,H) bf16
{
    const uint32_t row = blockIdx.x;              // (c*E + e)*MAX_TOK + t
    const uint32_t t   = row & (MAX_TOK - 1);
    const uint32_t ce  = row >> 11;               // MAX_TOK == 2048

    // Uniform across the block: scalar load + scalar branch, whole block exits.
    if ((int)t >= counter[ce]) return;

    const size_t mbase = (size_t)row * 3;
    const int chip = meta[mbase + 0];
    const int tok  = meta[mbase + 1];
    const int kk   = meta[mbase + 2];

    const size_t src = (size_t)row * HIDDEN;
    const size_t dst = (((size_t)chip * SEQ_LEN_PER_CHIP + (size_t)tok)
                        * NUM_EXPERTS_PER_TOK + (size_t)kk) * HIDDEN;

    // 256 threads * 8 bf16 = 2048 = HIDDEN. Non-temporal: the source is
    // streamed exactly once; keep L2 capacity for the output RMW lines.
    const v4u* p = (const v4u*)(disp + src) + threadIdx.x;
    v4u q = __builtin_nontemporal_load(p);

    // One 64-bit address; four packed-bf16 atomics via immediate offsets.
    uint32_t* ow = (uint32_t*)(out + dst) + (size_t)threadIdx.x * 4;
    asm volatile(
        "global_atomic_pk_add_bf16 %0, %1, off\n\t"
        "global_atomic_pk_add_bf16 %0, %2, off offset:4\n\t"
        "global_atomic_pk_add_bf16 %0, %3, off offset:8\n\t"
        "global_atomic_pk_add_bf16 %0, %4, off offset:12"
        :
        : "v"(ow), "v"(q.x), "v"(q.y), "v"(q.z), "v"(q.w)
        : "memory");
}

// ---------------------------------------------------------------------------
// Zero pass: clear the 512 MiB output with 128-bit stores. Must run every
// launch: the scatter pass accumulates with atomics and the harness does not
// re-poison between replays.
// ---------------------------------------------------------------------------
__global__ void __launch_bounds__(256) zero_kernel(v4u* __restrict__ out, long long n_vec4) {
    long long i = (long long)blockIdx.x * 256 + threadIdx.x;
    if (i < n_vec4) {
        v4u z = {0u, 0u, 0u, 0u};
        out[i] = z;
    }
}

extern "C" void kernel_launch(void* const* d_in, const int* in_sizes, int n_in,
                              void* d_out, int out_size, void* d_ws, size_t ws_size,
                              hipStream_t stream) {
    (void)in_sizes; (void)n_in; (void)d_ws; (void)ws_size;

    const __hip_bfloat16* disp    = (const __hip_bfloat16*)d_in[0];
    const int*            meta    = (const int*)d_in[1];
    const int*            counter = (const int*)d_in[2];
    __hip_bfloat16*       out     = (__hip_bfloat16*)d_out;

    // Zero the output (out_size bf16 elements; 8 per 128-bit vector).
    const long long n_vec4 = (long long)out_size / 8;
    const int zgrid = (int)((n_vec4 + 255) / 256);
    zero_kernel<<<zgrid, 256, 0, stream>>>((v4u*)d_out, n_vec4);

    // One block per source row; stream order serializes zero -> scatter.
    const int rows = NUM_CHIPS * EXPERTS_PER_CHIP * MAX_TOK;  // 131072
    combine_scatter<<<rows, 256, 0, stream>>>(disp, meta, counter, out);
}